// SpringSimulation_42374147342976
// MI455X (gfx1250) — compile-verified
//
#include <hip/hip_runtime.h>
#include <hip/hip_bf16.h>
#include <math.h>

#define TPB    256
#define UNROLL 10
#define TSTEP  0.01f

// CDNA5 async global->LDS path (gfx1250). Guarded so the file still compiles
// cleanly if a toolchain lacks the builtins; fallback uses direct b128 loads
// plus global_prefetch_b8.
#if defined(__HIP_DEVICE_COMPILE__) && \
    __has_builtin(__builtin_amdgcn_global_load_async_to_lds_b128) && \
    __has_builtin(__builtin_amdgcn_s_wait_asynccnt)
#define HAS_ASYNC_LDS 1
#else
#define HAS_ASYNC_LDS 0
#endif

typedef int v4i __attribute__((vector_size(16)));
typedef __attribute__((address_space(1))) v4i GV4;  // global int4
typedef __attribute__((address_space(3))) v4i LV4;  // LDS int4

// ---------------------------------------------------------------------------
// Edge kernel: per edge, gather pos[src], pos[dst] (L2-resident), compute
// spring acceleration, scatter-add +/- into accel[] with native f32 atomics.
// Edge list (128 MB) is the HBM-bound stream -> stage it through LDS with
// CDNA5 async loads, double buffered per wave (each lane owns its 16B slot,
// so no workgroup barrier is needed).
// ---------------------------------------------------------------------------
__global__ __launch_bounds__(TPB) void spring_edge_kernel(
    const float* __restrict__ pos,
    const long long* __restrict__ edges,   // [num_edges][2] int64
    float* __restrict__ accel,
    long long num_edges)
{
#if HAS_ASYNC_LDS
  __shared__ alignas(16) long long sedges[2 * TPB * 2];  // 2 bufs * TPB edges * 16B = 8KB
#endif
  const int tid = threadIdx.x;
  const long long base = (long long)blockIdx.x * ((long long)TPB * UNROLL);

#if HAS_ASYNC_LDS
  {
    long long e0  = base + tid;
    long long ce0 = (e0 < num_edges) ? e0 : 0;  // clamp: harmless read for tail lanes
    __builtin_amdgcn_global_load_async_to_lds_b128(
        (GV4*)(edges + 2 * ce0), (LV4*)(&sedges[2 * tid]), 0, 0);
  }
#endif

  for (int t = 0; t < UNROLL; ++t) {
    const long long e = base + (long long)t * TPB + tid;
    long long s64, d64;
#if HAS_ASYNC_LDS
    const int cur = t & 1;
    if (t + 1 < UNROLL) {
      long long en  = e + TPB;
      long long cen = (en < num_edges) ? en : 0;
      __builtin_amdgcn_global_load_async_to_lds_b128(
          (GV4*)(edges + 2 * cen),
          (LV4*)(&sedges[2 * (((t + 1) & 1) * TPB + tid)]), 0, 0);
      __builtin_amdgcn_s_wait_asynccnt(1);   // current tile landed, next in flight
    } else {
      __builtin_amdgcn_s_wait_asynccnt(0);
    }
    __asm__ __volatile__("" ::: "memory");
    s64 = sedges[2 * (cur * TPB + tid) + 0];
    d64 = sedges[2 * (cur * TPB + tid) + 1];
#else
    long long ce = (e < num_edges) ? e : 0;
    __builtin_prefetch(edges + 2 * ce + 2 * TPB * UNROLL, 0, 1);  // global_prefetch_b8
    s64 = edges[2 * ce + 0];
    d64 = edges[2 * ce + 1];
#endif
    if (e < num_edges) {
      const int s = (int)s64;
      const int d = (int)d64;
      const float* ps = pos + 3 * (long long)s;
      const float* pd = pos + 3 * (long long)d;
      float dx = ps[0] - pd[0];
      float dy = ps[1] - pd[1];
      float dz = ps[2] - pd[2];
      float dist = sqrtf(dx * dx + dy * dy + dz * dz);
      float c = (dist - 1.0f) / dist;        // SPRING_CONSTANT*(dist-target)/dist
      float ax = dx * c, ay = dy * c, az = dz * c;
      float* as_ = accel + 3 * (long long)s;
      float* ad_ = accel + 3 * (long long)d;
      unsafeAtomicAdd(as_ + 0,  ax);         // global_atomic_add_f32, no return
      unsafeAtomicAdd(as_ + 1,  ay);
      unsafeAtomicAdd(as_ + 2,  az);
      unsafeAtomicAdd(ad_ + 0, -ax);
      unsafeAtomicAdd(ad_ + 1, -ay);
      unsafeAtomicAdd(ad_ + 2, -az);
    }
  }
}

// ---------------------------------------------------------------------------
// Integration: new_v = v + dt*a ; new_p = p + dt*new_v.
// accel was accumulated in the first n3 floats of d_out; each thread reads
// then overwrites only its own slots -> race free. float4 path for the
// (always-hit) n3 % 4 == 0 case.
// ---------------------------------------------------------------------------
__global__ void spring_integrate4(const float4* __restrict__ pos,
                                  const float4* __restrict__ vel,
                                  float4* __restrict__ out_pos,   // holds accel on entry
                                  float4* __restrict__ out_vel,
                                  long long n4)
{
  long long i = (long long)blockIdx.x * blockDim.x + threadIdx.x;
  if (i < n4) {
    float4 a = out_pos[i];
    float4 v = vel[i];
    float4 p = pos[i];
    v.x += TSTEP * a.x;  v.y += TSTEP * a.y;  v.z += TSTEP * a.z;  v.w += TSTEP * a.w;
    p.x += TSTEP * v.x;  p.y += TSTEP * v.y;  p.z += TSTEP * v.z;  p.w += TSTEP * v.w;
    out_pos[i] = p;
    out_vel[i] = v;
  }
}

__global__ void spring_integrate1(const float* __restrict__ pos,
                                  const float* __restrict__ vel,
                                  float* __restrict__ out,
                                  long long n3)
{
  long long i = (long long)blockIdx.x * blockDim.x + threadIdx.x;
  if (i < n3) {
    float a = out[i];
    float v = vel[i] + TSTEP * a;
    float p = pos[i] + TSTEP * v;
    out[i]      = p;
    out[n3 + i] = v;
  }
}

extern "C" void kernel_launch(void* const* d_in, const int* in_sizes, int n_in,
                              void* d_out, int out_size, void* d_ws, size_t ws_size,
                              hipStream_t stream) {
  const float*     pos   = (const float*)d_in[0];
  const float*     vel   = (const float*)d_in[1];
  const long long* edges = (const long long*)d_in[2];  // int64 per reference
  const long long  n3        = (long long)in_sizes[0];       // 3 * num_points
  const long long  num_edges = (long long)in_sizes[2] / 2;
  float* out = (float*)d_out;

  // 1) zero the acceleration accumulator (first n3 floats of d_out)
  (void)hipMemsetAsync(out, 0, (size_t)n3 * sizeof(float), stream);

  // 2) edge scatter-add
  const long long per_block = (long long)TPB * UNROLL;
  int eblocks = (int)((num_edges + per_block - 1) / per_block);
  spring_edge_kernel<<<eblocks, TPB, 0, stream>>>(pos, edges, out, num_edges);

  // 3) integrate
  if ((n3 & 3) == 0) {
    long long n4 = n3 >> 2;
    int ib = (int)((n4 + TPB - 1) / TPB);
    spring_integrate4<<<ib, TPB, 0, stream>>>(
        (const float4*)pos, (const float4*)vel,
        (float4*)out, (float4*)(out + n3), n4);
  } else {
    int ib = (int)((n3 + TPB - 1) / TPB);
    spring_integrate1<<<ib, TPB, 0, stream>>>(pos, vel, out, n3);
  }
}